// torchPOS_12446815223815
// MI455X (gfx1250) — compile-verified
//
#include <hip/hip_runtime.h>
#include <hip/hip_bf16.h>

typedef float v2f __attribute__((ext_vector_type(2)));
typedef float v8f __attribute__((ext_vector_type(8)));

constexpr int TLEN  = 18000;         // samples per row
constexpr int BATCH = 16;            // rows
constexpr int WIN   = 48;            // window  (ceil(1.6*30))
constexpr int MWIN  = TLEN - WIN;    // 17952 windows actually overlap-added

// ---------------------------------------------------------------------------
// Stage 1: per-(b,n) window statistics -> per-window gains g_k[b,n]
//   g layout: g[k][n][b]  (b fastest, 16-wide => coalesced writes, and
//   matches the WMMA B-fragment gather in stage 2)
// ---------------------------------------------------------------------------
__global__ void pos_window_stats(const float* __restrict__ rgb,
                                 const float* __restrict__ matrix,
                                 float* __restrict__ g) {
  int tid = blockIdx.x * blockDim.x + threadIdx.x;
  if (tid >= BATCH * MWIN) return;
  int b = tid & 15;
  int n = tid >> 4;

  const float* p = rgb + (size_t)(b * TLEN + n) * 3;
  float s0 = 0.f, s1 = 0.f, s2 = 0.f;
  float q00 = 0.f, q01 = 0.f, q02 = 0.f, q11 = 0.f, q12 = 0.f, q22 = 0.f;
#pragma unroll 4
  for (int w = 0; w < WIN; ++w) {
    float r0 = p[3 * w + 0];
    float r1 = p[3 * w + 1];
    float r2 = p[3 * w + 2];
    s0 += r0; s1 += r1; s2 += r2;
    q00 = fmaf(r0, r0, q00); q01 = fmaf(r0, r1, q01); q02 = fmaf(r0, r2, q02);
    q11 = fmaf(r1, r1, q11); q12 = fmaf(r1, r2, q12); q22 = fmaf(r2, r2, q22);
  }

  const float invW = 1.0f / (float)WIN;
  float inv0 = 1.0f / (s0 * invW + 1e-6f);
  float inv1 = 1.0f / (s1 * invW + 1e-6f);
  float inv2 = 1.0f / (s2 * invW + 1e-6f);

  float m00 = matrix[0], m01 = matrix[1], m02 = matrix[2];
  float m10 = matrix[3], m11 = matrix[4], m12 = matrix[5];

  // std of S0 = sum_j m0j*inv_j*r_j over window (ddof=1)
  float u0 = m00 * inv0, u1 = m01 * inv1, u2 = m02 * inv2;
  float sum0 = u0 * s0 + u1 * s1 + u2 * s2;
  float sq0  = u0 * u0 * q00 + u1 * u1 * q11 + u2 * u2 * q22 +
               2.f * (u0 * u1 * q01 + u0 * u2 * q02 + u1 * u2 * q12);
  float var0 = (sq0 - sum0 * sum0 * invW) * (1.0f / (float)(WIN - 1));
  float std0 = sqrtf(fmaxf(var0, 0.f));

  float v0 = m10 * inv0, v1 = m11 * inv1, v2 = m12 * inv2;
  float sum1 = v0 * s0 + v1 * s1 + v2 * s2;
  float sq1  = v0 * v0 * q00 + v1 * v1 * q11 + v2 * v2 * q22 +
               2.f * (v0 * v1 * q01 + v0 * v2 * q02 + v1 * v2 * q12);
  float var1 = (sq1 - sum1 * sum1 * invW) * (1.0f / (float)(WIN - 1));
  float std1 = sqrtf(fmaxf(var1, 0.f));

  float c = std0 / (std1 + 1e-4f);

  // h[n,w] = sum_k r_k[n+w] * (m0k + c*m1k) * inv_k  ->  gains per window
  g[(size_t)(0 * MWIN + n) * 16 + b] = (m00 + c * m10) * inv0;
  g[(size_t)(1 * MWIN + n) * 16 + b] = (m01 + c * m11) * inv1;
  g[(size_t)(2 * MWIN + n) * 16 + b] = (m02 + c * m12) * inv2;
}

// ---------------------------------------------------------------------------
// Stage 2: overlap-add as a banded 16x64 (0/1) x 64x16 GEMM per t-tile using
// V_WMMA_F32_16X16X4_F32 accumulation (16 K-steps per series).
//   D[i][b] = sum_{j=i..i+47} g_k[b][t0-47+j]  == sliding sum A_k[b][t0+i]
//   ppg[t][b] = r0*A0 + r1*A1 + r2*A2   (ppg stored [t][b] for stage 3)
// One wave (32 lanes) per 16-sample tile. EXEC stays all-ones (no divergent
// branches: out-of-range g reads are clamped-address loads times a 0/1 mask).
// ---------------------------------------------------------------------------
__global__ void pos_overlap_add_wmma(const float* __restrict__ rgb,
                                     const float* __restrict__ g,
                                     float* __restrict__ ppg) {
  const int t0   = blockIdx.x * 16;
  const int lane = threadIdx.x;   // 0..31
  const int col  = lane & 15;     // N column == batch row; also A-matrix M row
  const int hi   = lane >> 4;     // half-wave select (K pairs / upper M half)
  const int base = t0 - 47;       // n = base + j, j in 0..63 (j=63 never hits)

  v8f acc[3];
#pragma unroll
  for (int k = 0; k < 3; ++k) {
    const float* gk = g + (size_t)k * MWIN * 16;
    v8f d = {0.f, 0.f, 0.f, 0.f, 0.f, 0.f, 0.f, 0.f};
#pragma unroll
    for (int s = 0; s < 16; ++s) {
      // A fragment (16x4 f32): lanes 0-15 hold K=4s+0/1, lanes 16-31 K=4s+2/3
      int j0 = 4 * s + 2 * hi;
      int j1 = j0 + 1;
      v2f a;
      a.x = (j0 >= col && j0 <= col + 47) ? 1.0f : 0.0f;
      a.y = (j1 >= col && j1 <= col + 47) ? 1.0f : 0.0f;
      // B fragment (4x16 f32): same half-wave K striping, N = col
      int n0 = base + j0;
      int n1 = base + j1;
      int c0 = min(max(n0, 0), MWIN - 1);
      int c1 = min(max(n1, 0), MWIN - 1);
      float ok0 = ((unsigned)n0 < (unsigned)MWIN) ? 1.0f : 0.0f;
      float ok1 = ((unsigned)n1 < (unsigned)MWIN) ? 1.0f : 0.0f;
      v2f bb;
      bb.x = gk[(size_t)c0 * 16 + col] * ok0;
      bb.y = gk[(size_t)c1 * 16 + col] * ok1;
      d = __builtin_amdgcn_wmma_f32_16x16x4_f32(
          /*neg_a=*/false, a, /*neg_b=*/false, bb,
          /*c_mod=*/(short)0, d, /*reuse_a=*/false, /*reuse_b=*/false);
    }
    acc[k] = d;
  }

  // D: vgpr p, lanes 0-15 -> i=p, lanes 16-31 -> i=p+8 ; i = t offset in tile
#pragma unroll
  for (int p = 0; p < 8; ++p) {
    int t = t0 + p + 8 * hi;
    const float* rp = rgb + (size_t)(col * TLEN + t) * 3;
    float v = rp[0] * acc[0][p] + rp[1] * acc[1][p] + rp[2] * acc[2][p];
    ppg[(size_t)t * 16 + col] = v;   // [t][b] layout, coalesced per half-wave
  }
}

// ---------------------------------------------------------------------------
// Stage 3: zero-state filtfilt (DF2T biquad forward, then on reversed signal).
// 16 independent rows -> 16 threads, sequential scans (inherently serial).
// ---------------------------------------------------------------------------
__global__ void pos_filtfilt(const float* __restrict__ ppg,
                             const float* __restrict__ acoef,
                             const float* __restrict__ bcoef,
                             float* __restrict__ ybuf,
                             float* __restrict__ out) {
  int b = threadIdx.x;
  if (b >= BATCH) return;
  float ia = 1.0f / acoef[0];
  float b0 = bcoef[0] * ia, b1 = bcoef[1] * ia, b2 = bcoef[2] * ia;
  float a1 = acoef[1] * ia, a2 = acoef[2] * ia;

  float z1 = 0.f, z2 = 0.f;
  for (int t = 0; t < TLEN; ++t) {
    float x = ppg[(size_t)t * 16 + b];
    float y = fmaf(b0, x, z1);
    z1 = fmaf(b1, x, fmaf(-a1, y, z2));
    z2 = fmaf(b2, x, -a2 * y);
    ybuf[(size_t)t * 16 + b] = y;
  }
  z1 = 0.f; z2 = 0.f;
  for (int k = 0; k < TLEN; ++k) {
    int t = TLEN - 1 - k;
    float x = ybuf[(size_t)t * 16 + b];
    float y = fmaf(b0, x, z1);
    z1 = fmaf(b1, x, fmaf(-a1, y, z2));
    z2 = fmaf(b2, x, -a2 * y);
    out[(size_t)b * TLEN + t] = y;   // output [B, T]
  }
}

// ---------------------------------------------------------------------------
extern "C" void kernel_launch(void* const* d_in, const int* in_sizes, int n_in,
                              void* d_out, int out_size, void* d_ws, size_t ws_size,
                              hipStream_t stream) {
  const float* rgb    = (const float*)d_in[0];  // [16,18000,3]
  const float* matrix = (const float*)d_in[1];  // [3,3]
  const float* acoef  = (const float*)d_in[2];  // [3]
  const float* bcoef  = (const float*)d_in[3];  // [3]
  float* out = (float*)d_out;                   // [16,18000]

  float* g    = (float*)d_ws;                   // 3*MWIN*16 floats
  float* ppg  = g + (size_t)3 * MWIN * 16;      // TLEN*16 floats
  float* ybuf = ppg + (size_t)TLEN * 16;        // TLEN*16 floats

  int nwork = BATCH * MWIN;
  pos_window_stats<<<(nwork + 255) / 256, 256, 0, stream>>>(rgb, matrix, g);
  pos_overlap_add_wmma<<<TLEN / 16, 32, 0, stream>>>(rgb, g, ppg);
  pos_filtfilt<<<1, 32, 0, stream>>>(ppg, acoef, bcoef, ybuf, out);
}